// GATModel_47725676593452
// MI455X (gfx1250) — compile-verified
//
#include <hip/hip_runtime.h>

// GAT (2-layer, heads=1) for gfx1250.
// GEMMs via V_WMMA_F32_16X16X4_F32 (fp32-exact matrix pipe).
// Edge stage: L2-resident gather/scatter with hw f32 atomics + ordered-uint atomic max.

typedef float v2f __attribute__((ext_vector_type(2)));
typedef float v8f __attribute__((ext_vector_type(8)));

#define HIDDEN 128

// ---------------------------------------------------------------------------
// GEMM: D[M,N] = A[M,K] @ B[K,N] (+ bias[N] if bias != null). fp32 WMMA.
// Block = 256 threads = 8 waves. A tile (BM x K) staged in LDS (row stride K+4
// -> conflict-free fragment reads). Each wave owns groups of 4 M-tiles sharing
// one N-tile, so each K-step loads the B fragment once and feeds 4 WMMAs.
// K, BM, N are compile-time so all hot addressing folds to immediates/shifts.
// ---------------------------------------------------------------------------
template <int K, int BM, int N>
__global__ __launch_bounds__(256) void gemm_f32_wmma(
    const float* __restrict__ A, const float* __restrict__ B,
    const float* __restrict__ bias, float* __restrict__ D, int M) {
  constexpr int KP = K + 4;
  constexpr int MT = 4;                    // M-tiles per wave-group
  __shared__ float As[BM * KP];
  const int m0 = blockIdx.x * BM;
  const int tid = threadIdx.x;

  // Cooperative, coalesced float4 load of the A tile; zero-pad rows >= M.
  for (int idx4 = tid; idx4 < BM * (K / 4); idx4 += 256) {
    const int r = idx4 / (K / 4);
    const int c = (idx4 % (K / 4)) * 4;
    float4 v = make_float4(0.f, 0.f, 0.f, 0.f);
    const int gr = m0 + r;
    if (gr < M) v = *reinterpret_cast<const float4*>(A + gr * K + c);
    *reinterpret_cast<float4*>(&As[r * KP + c]) = v;
  }
  __syncthreads();

  const int wave = tid >> 5;
  const int lane = tid & 31;
  const int half = lane >> 4;  // 0: lanes 0-15, 1: lanes 16-31
  const int l16 = lane & 15;
  const int kg = half * 2;     // K sub-offset per ISA A/B fragment layout

  constexpr int ntn = N >> 4;
  constexpr int ngroups = (BM >> 6) * ntn; // (BM/16/MT) * ntn
  const bool full = (m0 + BM) <= M;

  for (int grp = wave; grp < ngroups; grp += 8) {
    const int gm4 = grp / ntn;             // group of 4 M-tiles
    const int tn = grp % ntn;
    const float* Bp = B + tn * 16 + l16 + kg * N;  // rows k+kg, k+kg+1 at Bp[k*N], Bp[k*N+N]

    const float* Ar[MT];
    v8f acc[MT];
#pragma unroll
    for (int t = 0; t < MT; ++t) {
      Ar[t] = &As[((gm4 * MT + t) * 16 + l16) * KP + kg];
      acc[t] = 0.0f;
    }

#pragma unroll 4
    for (int k = 0; k < K; k += 4) {
      v2f bv;
      bv.x = Bp[k * N];                    // B[k+kg][n]
      bv.y = Bp[k * N + N];                // B[k+kg+1][n]
#pragma unroll
      for (int t = 0; t < MT; ++t) {
        const float2 af = *reinterpret_cast<const float2*>(Ar[t] + k);
        v2f av; av.x = af.x; av.y = af.y;  // A[m][k+kg], A[m][k+kg+1]
        acc[t] = __builtin_amdgcn_wmma_f32_16x16x4_f32(false, av, false, bv,
                                                       (short)0, acc[t], false, false);
      }
    }

    const int n_out = tn * 16 + l16;
    const float bval = bias ? bias[n_out] : 0.0f;
#pragma unroll
    for (int t = 0; t < MT; ++t) {
      const int mrow0 = m0 + (gm4 * MT + t) * 16 + half * 8;  // C/D: row = 8*half + r
      float* Dp = D + mrow0 * N + n_out;
      if (full) {
#pragma unroll
        for (int r = 0; r < 8; ++r) Dp[r * N] = acc[t][r] + bval;
      } else {
#pragma unroll
        for (int r = 0; r < 8; ++r)
          if (mrow0 + r < M) Dp[r * N] = acc[t][r] + bval;
      }
    }
  }
}

// ---------------------------------------------------------------------------
// Per-node attention coefficients: alpha_src[n] = dot(g[n,:], a_src), same dst.
// One wave per node; float4 per lane; cross-lane reduction via shfl_xor.
// ---------------------------------------------------------------------------
__global__ __launch_bounds__(256) void alpha_kernel(
    const float* __restrict__ g, const float* __restrict__ a_src,
    const float* __restrict__ a_dst, float* __restrict__ out_src,
    float* __restrict__ out_dst, int N) {
  const int node = (blockIdx.x * blockDim.x + threadIdx.x) >> 5;
  const int lane = threadIdx.x & 31;
  if (node >= N) return;
  const float4 gv = reinterpret_cast<const float4*>(g + node * HIDDEN)[lane];
  const float4 sv = reinterpret_cast<const float4*>(a_src)[lane];
  const float4 dv = reinterpret_cast<const float4*>(a_dst)[lane];
  float ps = gv.x * sv.x + gv.y * sv.y + gv.z * sv.z + gv.w * sv.w;
  float pd = gv.x * dv.x + gv.y * dv.y + gv.z * dv.z + gv.w * dv.w;
  for (int off = 16; off; off >>= 1) {
    ps += __shfl_xor(ps, off, 32);
    pd += __shfl_xor(pd, off, 32);
  }
  if (lane == 0) {
    out_src[node] = ps;
    out_dst[node] = pd;
  }
}

__global__ void fill_u32(unsigned* __restrict__ p, unsigned v, int n) {
  const int i = blockIdx.x * blockDim.x + threadIdx.x;
  if (i < n) p[i] = v;
}

// Monotonic float<->uint map so we can use hw global_atomic_max_u32 for f32 max.
__device__ __forceinline__ unsigned ord_encode(float f) {
  const unsigned u = __float_as_uint(f);
  return (u & 0x80000000u) ? ~u : (u | 0x80000000u);
}

__global__ void edge_pass1(const int* __restrict__ src, const int* __restrict__ dst,
                           const float* __restrict__ as_, const float* __restrict__ ad_,
                           float* __restrict__ e_arr, unsigned* __restrict__ m_u, int E) {
  const int i = blockIdx.x * blockDim.x + threadIdx.x;
  if (i >= E) return;
  const int s = src[i], d = dst[i];
  float e = as_[s] + ad_[d];
  e = (e > 0.f) ? e : 0.2f * e;  // leaky_relu, slope 0.2
  e_arr[i] = e;
  atomicMax(&m_u[d], ord_encode(e));
}

__global__ void fix_m(const unsigned* __restrict__ m_u, float* __restrict__ m_f, int N) {
  const int i = blockIdx.x * blockDim.x + threadIdx.x;
  if (i >= N) return;
  const unsigned u = m_u[i];
  float m = 0.f;  // empty segment (sentinel 0) -> 0, matching jnp.where(isfinite(m), m, 0)
  if (u != 0u) {
    const unsigned b = (u & 0x80000000u) ? (u ^ 0x80000000u) : ~u;
    if ((b & 0x7F800000u) != 0x7F800000u) m = __uint_as_float(b);
  }
  m_f[i] = m;
}

__global__ void edge_pass2(const int* __restrict__ dst, float* __restrict__ e_arr,
                           const float* __restrict__ m_f, float* __restrict__ ssum, int E) {
  const int i = blockIdx.x * blockDim.x + threadIdx.x;
  if (i >= E) return;
  const int d = dst[i];
  const float p = __expf(e_arr[i] - m_f[d]);
  e_arr[i] = p;
  __hip_atomic_fetch_add(&ssum[d], p, __ATOMIC_RELAXED, __HIP_MEMORY_SCOPE_AGENT);
}

// One wave per edge: gather g[src] (float4/lane), scale by alpha, scatter-add
// to acc[dst] with hw f32 atomics. Both arrays are L2-resident (51 MB each).
__global__ __launch_bounds__(256) void edge_pass3(
    const int* __restrict__ src, const int* __restrict__ dst,
    const float* __restrict__ p_arr, const float* __restrict__ ssum,
    const float* __restrict__ g, float* __restrict__ acc, int E) {
  const int edge = (blockIdx.x * blockDim.x + threadIdx.x) >> 5;
  const int lane = threadIdx.x & 31;
  if (edge >= E) return;
  const int s = src[edge], d = dst[edge];
  const float alpha = p_arr[edge] / (ssum[d] + 1e-16f);
  const float4 v = reinterpret_cast<const float4*>(g + s * HIDDEN)[lane];
  float* ap = acc + d * HIDDEN + lane * 4;
  __hip_atomic_fetch_add(ap + 0, v.x * alpha, __ATOMIC_RELAXED, __HIP_MEMORY_SCOPE_AGENT);
  __hip_atomic_fetch_add(ap + 1, v.y * alpha, __ATOMIC_RELAXED, __HIP_MEMORY_SCOPE_AGENT);
  __hip_atomic_fetch_add(ap + 2, v.z * alpha, __ATOMIC_RELAXED, __HIP_MEMORY_SCOPE_AGENT);
  __hip_atomic_fetch_add(ap + 3, v.w * alpha, __ATOMIC_RELAXED, __HIP_MEMORY_SCOPE_AGENT);
}

__global__ void finalize_relu(float* __restrict__ acc, const float* __restrict__ bias, int total) {
  const int i = blockIdx.x * blockDim.x + threadIdx.x;
  if (i >= total) return;
  const float v = acc[i] + bias[i & (HIDDEN - 1)];
  acc[i] = v > 0.f ? v : 0.f;
}

// ---------------------------------------------------------------------------
extern "C" void kernel_launch(void* const* d_in, const int* in_sizes, int n_in,
                              void* d_out, int out_size, void* d_ws, size_t ws_size,
                              hipStream_t stream) {
  const float* x     = (const float*)d_in[0];
  const int*   ei    = (const int*)d_in[1];
  const float* W_in  = (const float*)d_in[2];
  const float* b_in  = (const float*)d_in[3];
  const float* W1    = (const float*)d_in[4];
  const float* asr1  = (const float*)d_in[5];
  const float* ads1  = (const float*)d_in[6];
  const float* bias1 = (const float*)d_in[7];
  const float* W2    = (const float*)d_in[8];
  const float* asr2  = (const float*)d_in[9];
  const float* ads2  = (const float*)d_in[10];
  const float* bias2 = (const float*)d_in[11];
  const float* W_out = (const float*)d_in[12];
  const float* b_out = (const float*)d_in[13];

  const int N = in_sizes[0] / 256;  // IN_F = 256
  const int E = in_sizes[1] / 2;
  const int* src = ei;      // edge_index[0]
  const int* dst = ei + E;  // edge_index[1]

  // Workspace layout (two ping-pong node buffers + per-node/per-edge scalars).
  char* w = (char*)d_ws;
  float*    h    = (float*)w;    w += (size_t)N * HIDDEN * sizeof(float);
  float*    g    = (float*)w;    w += (size_t)N * HIDDEN * sizeof(float);
  float*    as_  = (float*)w;    w += (size_t)N * sizeof(float);
  float*    ad_  = (float*)w;    w += (size_t)N * sizeof(float);
  unsigned* m_u  = (unsigned*)w; w += (size_t)N * sizeof(unsigned);
  float*    m_f  = (float*)w;    w += (size_t)N * sizeof(float);
  float*    ssum = (float*)w;    w += (size_t)N * sizeof(float);
  float*    earr = (float*)w;    w += (size_t)E * sizeof(float);
  (void)ws_size; (void)n_in; (void)out_size;

  auto cdiv = [](long a, long b) { return (int)((a + b - 1) / b); };
  const int NH = N * HIDDEN;

  // h = x @ W_in + b_in
  gemm_f32_wmma<256, 64, 128><<<cdiv(N, 64), 256, 0, stream>>>(x, W_in, b_in, h, N);

  const float* Ws[2]  = {W1, W2};
  const float* Asv[2] = {asr1, asr2};
  const float* Adv[2] = {ads1, ads2};
  const float* Bs[2]  = {bias1, bias2};

  for (int L = 0; L < 2; ++L) {
    // g = h @ W   (h is dead afterwards -> reused as the scatter accumulator)
    gemm_f32_wmma<128, 128, 128><<<cdiv(N, 128), 256, 0, stream>>>(h, Ws[L], nullptr, g, N);
    alpha_kernel<<<cdiv((long)N * 32, 256), 256, 0, stream>>>(g, Asv[L], Adv[L], as_, ad_, N);
    fill_u32<<<cdiv(N, 256), 256, 0, stream>>>(m_u, 0u, N);
    fill_u32<<<cdiv(N, 256), 256, 0, stream>>>((unsigned*)ssum, 0u, N);
    fill_u32<<<cdiv(NH, 256), 256, 0, stream>>>((unsigned*)h, 0u, NH);
    edge_pass1<<<cdiv(E, 256), 256, 0, stream>>>(src, dst, as_, ad_, earr, m_u, E);
    fix_m<<<cdiv(N, 256), 256, 0, stream>>>(m_u, m_f, N);
    edge_pass2<<<cdiv(E, 256), 256, 0, stream>>>(dst, earr, m_f, ssum, E);
    edge_pass3<<<cdiv((long)E * 32, 256), 256, 0, stream>>>(src, dst, earr, ssum, g, h, E);
    finalize_relu<<<cdiv(NH, 256), 256, 0, stream>>>(h, Bs[L], NH);
  }

  // out = h @ W_out + b_out
  gemm_f32_wmma<128, 128, 64><<<cdiv(N, 128), 256, 0, stream>>>(h, W_out, b_out, (float*)d_out, N);
}